// YOLOv8Detector_11682311045677
// MI455X (gfx1250) — compile-verified
//
#include <hip/hip_runtime.h>
#include <stdint.h>

// ---------------- problem constants ----------------
#define B_       16
#define A_       8400
#define NC_      80
#define CONF_THR 0.5f
#define IOU_THR  0.45f
#define TOP_K_   1000
#define PADK     1024           // padded candidate count (32 u32 words of mask)
#define MAX_DET_ 300
#define MAX_WH_  7680.0f
#define ROWPAD   1008           // 63 row tiles of 16

typedef float    v2f __attribute__((ext_vector_type(2)));
typedef float    v8f __attribute__((ext_vector_type(8)));
typedef uint32_t v4u __attribute__((ext_vector_type(4)));
typedef int      v8i __attribute__((ext_vector_type(8)));
typedef int      v4i __attribute__((ext_vector_type(4)));

// =====================================================================
// Kernel 1: per-anchor class max/argmax + cxcywh -> xyxy decode.
// 43 MB streamed once: this is the HBM-roofline kernel (~2us at 23.3TB/s).
// =====================================================================
__global__ void __launch_bounds__(256)
decode_kernel(const float* __restrict__ boxes, const float* __restrict__ cls,
              float* __restrict__ dec_score, int* __restrict__ dec_cls,
              float* __restrict__ dec_box) {
  int n = blockIdx.x * blockDim.x + threadIdx.x;
  if (n >= B_ * A_) return;
  const float4* c = (const float4*)(cls + (size_t)n * NC_);
  float best = -1.0f; int bi = 0;
#pragma unroll
  for (int k = 0; k < NC_ / 4; ++k) {
    float4 v = c[k];
    if (v.x > best) { best = v.x; bi = 4 * k + 0; }
    if (v.y > best) { best = v.y; bi = 4 * k + 1; }
    if (v.z > best) { best = v.z; bi = 4 * k + 2; }
    if (v.w > best) { best = v.w; bi = 4 * k + 3; }
  }
  dec_score[n] = best;
  dec_cls[n]   = bi;
  float4 bx = ((const float4*)boxes)[n];
  float4 o;
  o.x = bx.x - 0.5f * bx.z; o.y = bx.y - 0.5f * bx.w;
  o.z = bx.x + 0.5f * bx.z; o.w = bx.y + 0.5f * bx.w;
  ((float4*)dec_box)[n] = o;
}

// =====================================================================
// Kernel 2: per-batch exact top-1000 via 32-bit radix-select on float
// bits (monotonic for positives), then bitonic sort of <=1024 survivors.
// One 256-thread block per batch; everything in LDS (~42 KB of 320 KB).
// =====================================================================
__global__ void __launch_bounds__(256)
topk_kernel(const float* __restrict__ dec_score, const int* __restrict__ dec_cls,
            const float* __restrict__ dec_box,
            float* __restrict__ top_score, int* __restrict__ top_cls,
            float* __restrict__ top_box_off, float* __restrict__ top_box_raw) {
  __shared__ uint32_t skeys[A_];
  __shared__ uint32_t skey[PADK];
  __shared__ int      sidx[PADK];
  __shared__ int      s_cnt, s_k;
  __shared__ uint32_t s_prefix;

  int b = blockIdx.x, tid = threadIdx.x;
  const float* sc = dec_score + (size_t)b * A_;

  for (int i = tid; i < A_; i += 256) {
    float s = sc[i];
    skeys[i] = (s > CONF_THR) ? __float_as_uint(s) : 0u;   // invalid -> key 0
  }
  if (tid == 0) { s_prefix = 0u; s_k = TOP_K_; }
  __syncthreads();

  // radix-select: prefix ends as the k-th largest key
  for (int bit = 31; bit >= 0; --bit) {
    if (tid == 0) s_cnt = 0;
    __syncthreads();
    uint32_t cand = (s_prefix | (1u << bit)) >> bit;
    int local = 0;
    for (int i = tid; i < A_; i += 256)
      if ((skeys[i] >> bit) == cand) ++local;
    atomicAdd(&s_cnt, local);
    __syncthreads();
    if (tid == 0) {
      if (s_cnt >= s_k) s_prefix |= (1u << bit);
      else              s_k -= s_cnt;
    }
    __syncthreads();
  }
  uint32_t thrk = s_prefix;

  for (int i = tid; i < PADK; i += 256) { skey[i] = 0u; sidx[i] = -1; }
  if (tid == 0) s_cnt = 0;
  __syncthreads();
  for (int i = tid; i < A_; i += 256) {
    uint32_t kk = skeys[i];
    if (kk > 0u && kk >= thrk) {
      int p = atomicAdd(&s_cnt, 1);
      if (p < PADK) { skey[p] = kk; sidx[p] = i; }
    }
  }
  __syncthreads();

  // bitonic sort, descending by key
  for (int k = 2; k <= PADK; k <<= 1) {
    for (int j = k >> 1; j > 0; j >>= 1) {
      for (int t = tid; t < PADK; t += 256) {
        int ixj = t ^ j;
        if (ixj > t) {
          bool up = ((t & k) == 0);
          uint32_t a = skey[t], c = skey[ixj];
          if (up ? (a < c) : (a > c)) {
            skey[t] = c; skey[ixj] = a;
            int ia = sidx[t]; sidx[t] = sidx[ixj]; sidx[ixj] = ia;
          }
        }
      }
      __syncthreads();
    }
  }

  // emit padded candidate arrays (sentinel boxes have zero area, far away)
  for (int t = tid; t < PADK; t += 256) {
    uint32_t kk = skey[t];
    size_t o = (size_t)b * PADK + t;
    if (kk > 0u && t < TOP_K_) {
      int idx = sidx[t];
      float  s   = __uint_as_float(kk);
      int    c   = dec_cls[(size_t)b * A_ + idx];
      float4 raw = ((const float4*)dec_box)[(size_t)b * A_ + idx];
      float  off = (float)c * MAX_WH_;
      top_score[o] = s;
      top_cls[o]   = c;
      ((float4*)top_box_raw)[o] = raw;
      ((float4*)top_box_off)[o] = make_float4(raw.x + off, raw.y + off,
                                              raw.z + off, raw.w + off);
    } else {
      float v = -1.0e7f - (float)t * 64.0f;
      top_score[o] = -1.0f;
      top_cls[o]   = -1;
      ((float4*)top_box_raw)[o] = make_float4(0.f, 0.f, 0.f, 0.f);
      ((float4*)top_box_off)[o] = make_float4(v, v, v, v);   // zero-area, unique
    }
  }
}

// ---------------------------------------------------------------------
// TDM helper: DMA a contiguous run of `nElem` f32 from global to LDS.
// D# bitfields per CDNA5 ISA ch.8 (group0: count/lds_addr/global_addr/type,
// group1: data_size / tensor_dim0 / tensor_dim1=1 / tile_dim0 / tile_dim1=1
// / tensor_dim0_stride). This toolchain exposes the 6-arg builtin:
// (v4u g0, v8i g1, v4i g2, v4i g3, v8i g4, i32 cpol).
// ---------------------------------------------------------------------
#if defined(__HIP_DEVICE_COMPILE__) && defined(__gfx1250__)
__device__ __forceinline__ void tdm_load_f32(uint32_t lds_off, const float* gptr,
                                             uint32_t nElem) {
  uint64_t ga = (uint64_t)(uintptr_t)gptr;
  v4u g0;
  g0.x = 1u;                                          // count=1 (valid D#)
  g0.y = lds_off;                                     // lds_addr [63:32]
  g0.z = (uint32_t)(ga & 0xffffffffu);                // global_addr [95:64]
  g0.w = (uint32_t)((ga >> 32) & 0x1ffffffu)          // global_addr [120:96]
       | (2u << 30);                                  // type=2 ("image")
  v8i g1;
  g1[0] = (int)(2u << 16);                            // data_size=2 (4 bytes)
  g1[1] = (int)((nElem & 0xffffu) << 16);             // tensor_dim0[15:0] @ bits 63:48
  g1[2] = (int)(((nElem >> 16) & 0xffffu)             // tensor_dim0[31:16]
              | (1u << 16));                          // tensor_dim1=1 (low half)
  g1[3] = (int)((nElem & 0xffffu) << 16);             // tile_dim0 @ bits 127:112
  g1[4] = 1;                                          // tile_dim1=1, tile_dim2=0
  g1[5] = (int)nElem;                                 // tensor_dim0_stride[31:0]
  g1[6] = 0;                                          // stride hi / dim1_stride
  g1[7] = 0;
  v4i z4 = {0, 0, 0, 0};
  v8i z8 = {0, 0, 0, 0, 0, 0, 0, 0};
  __builtin_amdgcn_tensor_load_to_lds(g0, g1, z4, z4, z8, 0);
}
#endif

// =====================================================================
// Kernel 3: suppression bitmask. One wave32 per 16x32 tile of the
// 1000x1000 pair matrix. TDM stages the 16 row / 32 col boxes into LDS;
// V_WMMA_F32_16X16X4_F32 computes the (area_i + area_j) outer-sum tile
// (A=[area_row,1,0,0], B=[1;area_col;0;0]); each lane finishes 16 IoUs
// against its fixed column box; ballot_w32 packs iou>thr bits; lanes
// 0..15 store one full u32 mask word each (no atomics needed).
// =====================================================================
__global__ void __launch_bounds__(32)
nms_mask_kernel(const float* __restrict__ top_box_off, uint32_t* __restrict__ mask) {
  int jt   = blockIdx.x;          // row tile   (0..62) -> rows jt*16..+15
  int ic   = blockIdx.y;          // col chunk  (0..31) -> cols ic*32..+31
  int b    = blockIdx.z;
  int lane = threadIdx.x;

  __shared__ float    rowbox[16 * 4];
  __shared__ float    colbox[32 * 4];
  __shared__ uint32_t balbuf[16];

  const float* rbase = top_box_off + ((size_t)b * PADK + (size_t)jt * 16) * 4;
  const float* cbase = top_box_off + ((size_t)b * PADK + (size_t)ic * 32) * 4;

#if defined(__HIP_DEVICE_COMPILE__) && defined(__gfx1250__)
  tdm_load_f32((uint32_t)(uintptr_t)rowbox, rbase, 16u * 4u);
  tdm_load_f32((uint32_t)(uintptr_t)colbox, cbase, 32u * 4u);
  __builtin_amdgcn_s_wait_tensorcnt(0);               // single wave: done
#else
  if (lane < 16) ((float4*)rowbox)[lane] = ((const float4*)rbase)[lane];
  ((float4*)colbox)[lane] = ((const float4*)cbase)[lane];
#endif
  __syncthreads();

  int n = lane & 15;                                  // column within tile (fixed/lane)
  float4 cb0 = ((float4*)colbox)[n];
  float4 cb1 = ((float4*)colbox)[n + 16];
  float carea0 = (cb0.z - cb0.x) * (cb0.w - cb0.y);
  float carea1 = (cb1.z - cb1.x) * (cb1.w - cb1.y);

  float4 ra = ((float4*)rowbox)[lane & 15];           // no divergence
  float rarea = (ra.z - ra.x) * (ra.w - ra.y);
  bool  lo = (lane < 16);

  v8f d0 = {}; v8f d1 = {};
#if defined(__HIP_DEVICE_COMPILE__) && defined(__gfx1250__)
  // A 16x4: lanes0-15 {K0=area_row, K1=1}, lanes16-31 {K2=0, K3=0}
  v2f Am; Am.x = lo ? rarea : 0.0f;  Am.y = lo ? 1.0f : 0.0f;
  // B 4x16: lanes0-15 {K0=1, K1=area_col}, lanes16-31 {K2=0, K3=0}
  v2f B0; B0.x = lo ? 1.0f : 0.0f;   B0.y = lo ? carea0 : 0.0f;
  v2f B1; B1.x = lo ? 1.0f : 0.0f;   B1.y = lo ? carea1 : 0.0f;
  v8f cz = {};
  d0 = __builtin_amdgcn_wmma_f32_16x16x4_f32(false, Am, false, B0,
                                             (short)0, cz, false, false);
  d1 = __builtin_amdgcn_wmma_f32_16x16x4_f32(false, Am, false, B1,
                                             (short)0, cz, false, false);
#else
  (void)rarea; (void)lo;
#endif

  // C/D layout: VGPR r, lane l -> M = r + (l>=16 ? 8:0), N = l&15
  int mbase = (lane >= 16) ? 8 : 0;
#pragma unroll
  for (int r = 0; r < 8; ++r) {
    float4 rb = ((float4*)rowbox)[mbase + r];
    float ix0 = fminf(rb.z, cb0.z) - fmaxf(rb.x, cb0.x);
    float iy0 = fminf(rb.w, cb0.w) - fmaxf(rb.y, cb0.y);
    float in0 = fmaxf(ix0, 0.0f) * fmaxf(iy0, 0.0f);
    float ix1 = fminf(rb.z, cb1.z) - fmaxf(rb.x, cb1.x);
    float iy1 = fminf(rb.w, cb1.w) - fmaxf(rb.y, cb1.y);
    float in1 = fmaxf(ix1, 0.0f) * fmaxf(iy1, 0.0f);
    // iou > thr  <=>  inter > thr*(areaSum - inter + eps)   (denom > 0)
    bool hit0 = in0 > IOU_THR * (d0[r] - in0 + 1e-7f);
    bool hit1 = in1 > IOU_THR * (d1[r] - in1 + 1e-7f);
    uint32_t m0 = __builtin_amdgcn_ballot_w32(hit0);
    uint32_t m1 = __builtin_amdgcn_ballot_w32(hit1);
    if (lane == 0) { balbuf[r] = m0; balbuf[8 + r] = m1; }
  }
  __syncthreads();

  if (lane < 16) {
    uint32_t h0, h1;
    if (lane < 8) { h0 = balbuf[lane] & 0xffffu;      h1 = balbuf[8 + lane] & 0xffffu; }
    else          { h0 = balbuf[lane - 8] >> 16;      h1 = balbuf[8 + lane - 8] >> 16; }
    int row = jt * 16 + lane;
    mask[((size_t)b * ROWPAD + row) * 32 + ic] = h0 | (h1 << 16);
  }
}

// =====================================================================
// Kernel 4: greedy scan. One wave32 per batch: lane = mask word index,
// so the 1024-bit "already suppressed" set lives in exactly 32 lanes.
// Then zero-fill + serial emit of the first 300 kept (already in score
// order because candidates were sorted descending).
// =====================================================================
__global__ void __launch_bounds__(32)
nms_scan_kernel(const uint32_t* __restrict__ mask, const float* __restrict__ top_score,
                const int* __restrict__ top_cls, const float* __restrict__ top_box_raw,
                float* __restrict__ out_boxes, float* __restrict__ out_scores,
                int* __restrict__ out_cls) {
  int b = blockIdx.x, lane = threadIdx.x;
  volatile __shared__ uint32_t accum[32];
  __shared__ float   ssc[PADK];
  __shared__ uint8_t keepf[TOP_K_];

  accum[lane] = 0u;
  for (int i = lane; i < PADK; i += 32)  ssc[i] = top_score[(size_t)b * PADK + i];
  for (int i = lane; i < TOP_K_; i += 32) keepf[i] = 0;
  __syncthreads();

  const uint32_t* mrow = mask + (size_t)b * ROWPAD * 32;
  for (int i = 0; i < TOP_K_; ++i) {
    uint32_t w = accum[i >> 5];                     // wave-uniform
    bool sup  = (w >> (i & 31)) & 1u;
    bool keep = (!sup) && (ssc[i] > CONF_THR);      // uniform branch
    if (keep) {
      accum[lane] |= mrow[(size_t)i * 32 + lane];
      if (lane == 0) keepf[i] = 1;
    }
  }
  __syncthreads();

  for (int s = lane; s < MAX_DET_; s += 32) {
    size_t o = (size_t)b * MAX_DET_ + s;
    ((float4*)out_boxes)[o] = make_float4(0.f, 0.f, 0.f, 0.f);
    out_scores[o] = 0.0f;
    out_cls[o]    = -1;
  }
  __syncthreads();

  if (lane == 0) {
    int out = 0;
    for (int i = 0; i < TOP_K_ && out < MAX_DET_; ++i) {
      if (keepf[i]) {
        size_t src = (size_t)b * PADK + i;
        size_t o   = (size_t)b * MAX_DET_ + out;
        ((float4*)out_boxes)[o] = ((const float4*)top_box_raw)[src];
        out_scores[o] = ssc[i];
        out_cls[o]    = top_cls[src];
        ++out;
      }
    }
  }
}

// =====================================================================
extern "C" void kernel_launch(void* const* d_in, const int* in_sizes, int n_in,
                              void* d_out, int out_size, void* d_ws, size_t ws_size,
                              hipStream_t stream) {
  (void)in_sizes; (void)n_in; (void)out_size; (void)ws_size;
  const float* boxes = (const float*)d_in[0];   // (B,A,4) cxcywh
  const float* cls   = (const float*)d_in[1];   // (B,A,NC)

  char* ws = (char*)d_ws;
  size_t off = 0;
  auto alloc = [&](size_t bytes) { size_t o = off; off += (bytes + 255) & ~(size_t)255; return o; };

  float*    dec_score   = (float*)(ws + alloc((size_t)B_ * A_ * 4));
  int*      dec_cls     = (int*)  (ws + alloc((size_t)B_ * A_ * 4));
  float*    dec_box     = (float*)(ws + alloc((size_t)B_ * A_ * 16));
  float*    top_score   = (float*)(ws + alloc((size_t)B_ * PADK * 4));
  int*      top_cls     = (int*)  (ws + alloc((size_t)B_ * PADK * 4));
  float*    top_box_off = (float*)(ws + alloc((size_t)B_ * PADK * 16));
  float*    top_box_raw = (float*)(ws + alloc((size_t)B_ * PADK * 16));
  uint32_t* mask        = (uint32_t*)(ws + alloc((size_t)B_ * ROWPAD * 32 * 4));

  // output: boxes (B,300,4) | scores (B,300) | cls int32 (B,300)
  float* out_boxes  = (float*)d_out;
  float* out_scores = out_boxes + (size_t)B_ * MAX_DET_ * 4;
  int*   out_cls    = (int*)(out_scores + (size_t)B_ * MAX_DET_);

  decode_kernel<<<(B_ * A_ + 255) / 256, 256, 0, stream>>>(
      boxes, cls, dec_score, dec_cls, dec_box);
  topk_kernel<<<B_, 256, 0, stream>>>(
      dec_score, dec_cls, dec_box, top_score, top_cls, top_box_off, top_box_raw);
  nms_mask_kernel<<<dim3(63, 32, B_), 32, 0, stream>>>(top_box_off, mask);
  nms_scan_kernel<<<B_, 32, 0, stream>>>(
      mask, top_score, top_cls, top_box_raw, out_boxes, out_scores, out_cls);
}